// GCN2Conv_67448166416642
// MI455X (gfx1250) — compile-verified
//
#include <hip/hip_runtime.h>

typedef float v2f __attribute__((ext_vector_type(2)));
typedef float v8f __attribute__((ext_vector_type(8)));

#define Bsz 4
#define Nn  3000
#define Ff  128
#define RT  ((Nn + 15) / 16)        // 188 row tiles per batch

static __device__ __forceinline__ v8f wmma4(v2f a, v2f b, v8f c) {
  return __builtin_amdgcn_wmma_f32_16x16x4_f32(false, a, false, b, (short)0, c,
                                               false, false);
}

// ---------------------------------------------------------------------------
// Kernel 1: deg = rowsum(adj)+1 ; dinv = rsqrt(deg) ;
//           xsT[b][c][i] = dinv * x[b][i][c]   (transposed for b64 B-loads)
// one 256-thread block per (b, i) row
// ---------------------------------------------------------------------------
__global__ void gcn_deg_scale(const float* __restrict__ adj,
                              const float* __restrict__ x,
                              float* __restrict__ dinv,
                              float* __restrict__ xsT) {
  int bi = blockIdx.x;                         // b*Nn + i
  int b = bi / Nn, i = bi % Nn;
  const float* row = adj + (size_t)bi * Nn;
  float s = 0.0f;
  for (int j = threadIdx.x; j < Nn; j += 256) s += row[j];
  __shared__ float red[256];
  red[threadIdx.x] = s;
  __syncthreads();
  for (int o = 128; o > 0; o >>= 1) {
    if (threadIdx.x < o) red[threadIdx.x] += red[threadIdx.x + o];
    __syncthreads();
  }
  float d = rsqrtf(red[0] + 1.0f);             // +1 for self loop
  if (threadIdx.x == 0) dinv[bi] = d;
  if (threadIdx.x < Ff) {
    xsT[(size_t)b * Ff * Nn + (size_t)threadIdx.x * Nn + i] =
        d * x[(size_t)bi * Ff + threadIdx.x];
  }
}

// ---------------------------------------------------------------------------
// Kernel 1b: thetaT[g][f] = theta[f][g]
// ---------------------------------------------------------------------------
__global__ void gcn_transpose_theta(const float* __restrict__ theta,
                                    float* __restrict__ thetaT) {
  int idx = blockIdx.x * 256 + threadIdx.x;    // 0 .. 16383
  int r = idx >> 7, c = idx & 127;
  thetaT[c * Ff + r] = theta[idx];
}

// ---------------------------------------------------------------------------
// Kernel 2: support = (1-a) * dinv_i * (adj @ (dinv.x) + dinv_i*x_i) + a * h0
// Each wave: 16 rows x 64 cols of C (4 WMMA accumulators), K-loop over 3000.
// WMMA f32 16x16x4 layouts:
//   A: lanes 0-15 row M=lane, VGPR{0,1}=K{k,k+1}; lanes 16-31 K{k+2,k+3}
//   B: VGPR j: lanes 0-15 (K=k+j, N=lane), lanes 16-31 (K=k+j+2, N=lane-16)
//      -> with xsT[col][k] each lane's pair is contiguous: one b64 load
//   C: VGPR v: lanes 0-15 (M=v, N=lane), lanes 16-31 (M=v+8, N=lane-16)
// ---------------------------------------------------------------------------
__global__ void gcn_agg_wmma(const float* __restrict__ adj,
                             const float* __restrict__ xsT,
                             const float* __restrict__ dinv,
                             const float* __restrict__ x,
                             const float* __restrict__ h0,
                             const float* __restrict__ alpha_p,
                             float* __restrict__ support) {
  int wv   = (int)((blockIdx.x * blockDim.x + threadIdx.x) >> 5);
  int lane = threadIdx.x & 31;
  int colhalf = wv & 1;                        // 0: cols 0-63, 1: cols 64-127
  int rt = (wv >> 1) % RT;
  int b  = (wv >> 1) / RT;
  if (b >= Bsz) return;                        // uniform per wave

  int r0  = rt * 16;
  int m   = lane & 15;
  int kh2 = (lane >> 4) * 2;                   // K sub-offset 0 or 2
  int rowc = min(r0 + m, Nn - 1);              // clamp tail rows (loads only)
  int c0 = colhalf * 64;
  const float* arow = adj + (size_t)b * Nn * Nn + (size_t)rowc * Nn + kh2;
  const float* bcol = xsT + (size_t)b * Ff * Nn + (size_t)(c0 + m) * Nn + kh2;

  v8f acc[4] = {};
  for (int k = 0; k < Nn; k += 4) {
    v2f a = *(const v2f*)(arow + k);           // global_load_b64
#pragma unroll
    for (int g = 0; g < 4; ++g) {
      v2f bb = *(const v2f*)(bcol + (size_t)(g * 16) * Nn + k);  // b64
      acc[g] = wmma4(a, bb, acc[g]);
    }
  }

  float alpha    = alpha_p[0];
  float om_alpha = 1.0f - alpha;
  int rbase = r0 + (lane >> 4) * 8;            // C-layout row base
#pragma unroll
  for (int v = 0; v < 8; ++v) {
    int orow = rbase + v;
    if (orow < Nn) {
      float di  = dinv[b * Nn + orow];
      size_t ro = ((size_t)b * Nn + orow) * Ff;
#pragma unroll
      for (int g = 0; g < 4; ++g) {
        int ocol = c0 + g * 16 + m;
        // self loop term: dinv_i * x_i[c]
        float hi = di * (acc[g][v] + di * x[ro + ocol]);
        support[ro + ocol] = om_alpha * hi + alpha * h0[ro + ocol];
      }
    }
  }
}

// ---------------------------------------------------------------------------
// Kernel 3: out = beta * (support @ theta) + (1-beta) * support  (in place)
// Flat [12000 x 128] rows (12000 % 16 == 0). Wave: 16 rows x 128 cols.
// Safe in place: each wave reads/writes only its own 16 rows.
// ---------------------------------------------------------------------------
__global__ void gcn_theta_wmma(const float* __restrict__ thetaT,
                               const float* __restrict__ lamda_p,
                               const int*   __restrict__ l_p,
                               float* __restrict__ sup) {
  int wv   = (int)((blockIdx.x * blockDim.x + threadIdx.x) >> 5);
  int lane = threadIdx.x & 31;
  const int RTOT = (Bsz * Nn) / 16;            // 750
  if (wv >= RTOT) return;                      // uniform per wave

  int r0  = wv * 16;
  int m   = lane & 15;
  int kh2 = (lane >> 4) * 2;
  const float* ar = sup    + (size_t)(r0 + m) * Ff + kh2;
  const float* bc = thetaT + (size_t)m * Ff + kh2;

  v8f acc[8] = {};
  for (int k = 0; k < Ff; k += 4) {
    v2f a = *(const v2f*)(ar + k);
#pragma unroll
    for (int g = 0; g < 8; ++g) {
      v2f bb = *(const v2f*)(bc + (size_t)(g * 16) * Ff + k);  // b64
      acc[g] = wmma4(a, bb, acc[g]);
    }
  }

  float beta = logf(lamda_p[0] / (float)l_p[0] + 1.0f);
  float omb  = 1.0f - beta;
  int rbase = r0 + (lane >> 4) * 8;
#pragma unroll
  for (int v = 0; v < 8; ++v) {
    size_t ro = (size_t)(rbase + v) * Ff;
#pragma unroll
    for (int g = 0; g < 8; ++g) {
      int ocol = g * 16 + m;
      float s  = sup[ro + ocol];
      sup[ro + ocol] = beta * acc[g][v] + omb * s;
    }
  }
}

// ---------------------------------------------------------------------------
extern "C" void kernel_launch(void* const* d_in, const int* in_sizes, int n_in,
                              void* d_out, int out_size, void* d_ws, size_t ws_size,
                              hipStream_t stream) {
  const float* x     = (const float*)d_in[0];   // [B,N,F]
  const float* adj   = (const float*)d_in[1];   // [B,N,N]
  const float* h0    = (const float*)d_in[2];   // [B,N,F]
  const float* theta = (const float*)d_in[3];   // [F,F]
  const float* lamda = (const float*)d_in[4];   // scalar
  const float* alpha = (const float*)d_in[5];   // scalar
  const int*   l     = (const int*)d_in[6];     // scalar

  float* out    = (float*)d_out;                // also holds `support`
  float* dinv   = (float*)d_ws;                 // 12000 floats
  float* xsT    = (float*)d_ws + 12032;         // 4*128*3000 floats
  float* thetaT = (float*)d_ws + 12032 + Bsz * Ff * Nn;  // 128*128 floats

  // 1) degree + transposed scaled features; theta transpose
  gcn_deg_scale<<<Bsz * Nn, 256, 0, stream>>>(adj, x, dinv, xsT);
  gcn_transpose_theta<<<(Ff * Ff) / 256, 256, 0, stream>>>(theta, thetaT);

  // 2) aggregation matmul (WMMA f32), fused normalize/self-loop/alpha-blend
  int waves2 = Bsz * RT * 2;                    // 1504 waves, 8 waves/block
  gcn_agg_wmma<<<waves2 / 8, 256, 0, stream>>>(adj, xsT, dinv, x, h0, alpha, out);

  // 3) theta matmul + beta residual, in place on d_out
  int waves3 = (((Bsz * Nn) / 16) + 7) & ~7;    // 752 (750 active)
  gcn_theta_wmma<<<waves3 / 8, 256, 0, stream>>>(thetaT, lamda, l, out);
}